// LargeAssociativeMem_37452114821968
// MI455X (gfx1250) — compile-verified
//
#include <hip/hip_runtime.h>

typedef __attribute__((ext_vector_type(2))) float v2f;
typedef __attribute__((ext_vector_type(4))) float v4f;
typedef __attribute__((ext_vector_type(8))) float v8f;

#define Bv 64
#define Mv 2048
#define Nv 16384
#define KT 32
#define LSTR 34            // LDS row stride in dwords (even -> 8B-aligned b64 reads)
#define NCHUNK 32
#define KC (Nv / NCHUNK)   // 512
#define DT_TAU 0.1f
#define FP_THRES 1e-3f

// ---------------------------------------------------------------- helpers
__device__ __forceinline__ void lds_put4(float* base, int row, int c, v4f d) {
  v2f* p = (v2f*)&base[row * LSTR + c];
  v2f t0; t0.x = d.x; t0.y = d.y;
  v2f t1; t1.x = d.z; t1.y = d.w;
  p[0] = t0; p[1] = t1;
}

// One KT-deep WMMA pass: A rows = 64 batch rows (4 subtiles), B col-subtile from LDS.
__device__ __forceinline__ void mm_lds_step(const float* __restrict__ Ab,
                                            const float* __restrict__ Bb,
                                            int wv, int lr, int lh, v8f acc[4]) {
#pragma unroll
  for (int k2 = 0; k2 < KT; k2 += 4) {
    int kk = k2 + 2 * lh;
    v2f bv = *(const v2f*)&Bb[(wv * 16 + lr) * LSTR + kk];
#pragma unroll
    for (int bt = 0; bt < 4; ++bt) {
      v2f av = *(const v2f*)&Ab[(bt * 16 + lr) * LSTR + kk];
      acc[bt] = __builtin_amdgcn_wmma_f32_16x16x4_f32(
          false, av, false, bv, (short)0, acc[bt], false, false);
    }
  }
}

// ---------------------------------------------------------------- init
__global__ __launch_bounds__(256) void k_init(const float* __restrict__ inp,
                                              float* __restrict__ v,
                                              int* __restrict__ conv) {
  int i = blockIdx.x * 256 + threadIdx.x;
  v[i] = inp[i];
  if (i == 0) *conv = 0;
}

// ------------------------------------------- GEMM1: H[b][n] = sum_m v[b][m]*W[n][m]
// grid.x = Nv/128.  8 waves; wave w owns n-subtile w (16 cols) x all 64 batch rows.
// Software-pipelined: double-buffered LDS, one barrier per K-tile.
__global__ __launch_bounds__(256) void k_gemm1(const float* __restrict__ v,
                                               const float* __restrict__ W,
                                               float* __restrict__ H,
                                               const int* __restrict__ conv) {
  if (*conv) return;
  __shared__ float Vs[2][64 * LSTR];
  __shared__ float Ws[2][128 * LSTR];
  const int tid = threadIdx.x;
  const int lane = tid & 31;
  const int wv = tid >> 5;
  const int n0 = blockIdx.x * 128;
  const int lr = lane & 15;
  const int lh = lane >> 4;

  const int srow = tid >> 3;         // staging row base (0..31), +32 per i
  const int sc = (tid & 7) << 2;     // staging col (0..28)

  v8f acc[4] = {};
  v4f rv[2], rw[4];

  // prologue: fetch tile 0 and stage into buffer 0
#pragma unroll
  for (int i = 0; i < 2; ++i)
    rv[i] = *(const v4f*)(v + (srow + 32 * i) * Mv + sc);
#pragma unroll
  for (int i = 0; i < 4; ++i)
    rw[i] = *(const v4f*)(W + (size_t)(n0 + srow + 32 * i) * Mv + sc);
#pragma unroll
  for (int i = 0; i < 2; ++i) lds_put4(Vs[0], srow + 32 * i, sc, rv[i]);
#pragma unroll
  for (int i = 0; i < 4; ++i) lds_put4(Ws[0], srow + 32 * i, sc, rw[i]);
  __syncthreads();

  const int NIT = Mv / KT;   // 64
  for (int it = 0; it < NIT; ++it) {
    const bool more = (it + 1) < NIT;
    const int kb = (it + 1) * KT;
    if (more) {              // prefetch tile it+1 (overlaps with WMMA below)
#pragma unroll
      for (int i = 0; i < 2; ++i)
        rv[i] = *(const v4f*)(v + (srow + 32 * i) * Mv + kb + sc);
#pragma unroll
      for (int i = 0; i < 4; ++i)
        rw[i] = *(const v4f*)(W + (size_t)(n0 + srow + 32 * i) * Mv + kb + sc);
    }
    mm_lds_step(Vs[it & 1], Ws[it & 1], wv, lr, lh, acc);
    if (more) {
#pragma unroll
      for (int i = 0; i < 2; ++i) lds_put4(Vs[(it + 1) & 1], srow + 32 * i, sc, rv[i]);
#pragma unroll
      for (int i = 0; i < 4; ++i) lds_put4(Ws[(it + 1) & 1], srow + 32 * i, sc, rw[i]);
    }
    __syncthreads();
  }

  const int ncol = n0 + wv * 16 + lr;
#pragma unroll
  for (int bt = 0; bt < 4; ++bt)
#pragma unroll
    for (int r = 0; r < 8; ++r) {
      int brow = bt * 16 + r + 8 * lh;
      H[(size_t)brow * Nv + ncol] = acc[bt][r];
    }
}

// --------------------------------------- row softmax stats: max_b, 1/sumexp_b
__global__ __launch_bounds__(256) void k_stats(const float* __restrict__ H,
                                               float* __restrict__ stats,
                                               const int* __restrict__ conv) {
  if (*conv) return;
  __shared__ float sm[256], ss[256];
  const int b = blockIdx.x;
  const float* row = H + (size_t)b * Nv;
  float m = -3.0e38f, s = 0.0f;
  for (int i = threadIdx.x; i < Nv; i += 256) {
    float h = row[i];
    float nm = fmaxf(m, h);
    s = s * __expf(m - nm) + __expf(h - nm);
    m = nm;
  }
  sm[threadIdx.x] = m; ss[threadIdx.x] = s;
  __syncthreads();
  for (int off = 128; off > 0; off >>= 1) {
    if (threadIdx.x < off) {
      float m1 = sm[threadIdx.x], s1 = ss[threadIdx.x];
      float m2 = sm[threadIdx.x + off], s2 = ss[threadIdx.x + off];
      float M = fmaxf(m1, m2);
      ss[threadIdx.x] = s1 * __expf(m1 - M) + s2 * __expf(m2 - M);
      sm[threadIdx.x] = M;
    }
    __syncthreads();
  }
  if (threadIdx.x == 0) { stats[b] = sm[0]; stats[64 + b] = 1.0f / ss[0]; }
}

// ---------- GEMM2 (split-K): parts[kc][b][m] = sum_{k in chunk} f[b][k]*W[k][m]
// grid = (Mv/128, NCHUNK). Wave w owns m-subtile w x all 64 batch rows.
// F tiles double-buffered in LDS (softmax applied at stage time); B operand read
// directly from L2-resident W with lanes coalesced along m.
__global__ __launch_bounds__(256) void k_gemm2(const float* __restrict__ H,
                                               const float* __restrict__ W,
                                               const float* __restrict__ stats,
                                               float* __restrict__ parts,
                                               const int* __restrict__ conv) {
  if (*conv) return;
  __shared__ float Fs[2][64 * LSTR];
  __shared__ float smax[64], srs[64];
  const int tid = threadIdx.x;
  if (tid < 64) { smax[tid] = stats[tid]; srs[tid] = stats[64 + tid]; }
  const int lane = tid & 31;
  const int wv = tid >> 5;
  const int m0 = blockIdx.x * 128;
  const int kc0 = blockIdx.y * KC;
  const int lr = lane & 15;
  const int lh = lane >> 4;

  const int srow = tid >> 3;
  const int sc = (tid & 7) << 2;
  const float* wcol = W + m0 + wv * 16 + lr;

  v8f acc[4] = {};
  v4f rh[2];

  // prologue
#pragma unroll
  for (int i = 0; i < 2; ++i)
    rh[i] = *(const v4f*)(H + (size_t)(srow + 32 * i) * Nv + kc0 + sc);
  __syncthreads();   // smax/srs ready
#pragma unroll
  for (int i = 0; i < 2; ++i) {
    int row = srow + 32 * i;
    float mx = smax[row], rs = srs[row];
    float* p = &Fs[0][row * LSTR + sc];
    p[0] = __expf(rh[i].x - mx) * rs;
    p[1] = __expf(rh[i].y - mx) * rs;
    p[2] = __expf(rh[i].z - mx) * rs;
    p[3] = __expf(rh[i].w - mx) * rs;
  }
  __syncthreads();

  const int NIT = KC / KT;   // 16
  for (int it = 0; it < NIT; ++it) {
    const bool more = (it + 1) < NIT;
    const int kb = (it + 1) * KT;
    if (more) {
#pragma unroll
      for (int i = 0; i < 2; ++i)
        rh[i] = *(const v4f*)(H + (size_t)(srow + 32 * i) * Nv + kc0 + kb + sc);
    }
    // compute: B from global (lanes along m -> coalesced, L2-resident)
    {
      const float* Fb = Fs[it & 1];
      const float* wbase = wcol + (size_t)(kc0 + it * KT) * Mv;
#pragma unroll
      for (int k2 = 0; k2 < KT; k2 += 4) {
        int kk = k2 + 2 * lh;
        const float* wp = wbase + (size_t)kk * Mv;
        v2f bv; bv.x = wp[0]; bv.y = wp[Mv];
#pragma unroll
        for (int bt = 0; bt < 4; ++bt) {
          v2f av = *(const v2f*)&Fb[(bt * 16 + lr) * LSTR + kk];
          acc[bt] = __builtin_amdgcn_wmma_f32_16x16x4_f32(
              false, av, false, bv, (short)0, acc[bt], false, false);
        }
      }
    }
    if (more) {
#pragma unroll
      for (int i = 0; i < 2; ++i) {
        int row = srow + 32 * i;
        float mx = smax[row], rs = srs[row];
        float* p = &Fs[(it + 1) & 1][row * LSTR + sc];
        p[0] = __expf(rh[i].x - mx) * rs;
        p[1] = __expf(rh[i].y - mx) * rs;
        p[2] = __expf(rh[i].z - mx) * rs;
        p[3] = __expf(rh[i].w - mx) * rs;
      }
    }
    __syncthreads();
  }

  float* P = parts + (size_t)blockIdx.y * (Bv * Mv) + m0 + wv * 16 + lr;
#pragma unroll
  for (int bt = 0; bt < 4; ++bt)
#pragma unroll
    for (int r = 0; r < 8; ++r) {
      int brow = bt * 16 + r + 8 * lh;
      P[(size_t)brow * Mv] = acc[bt][r];
    }
}

// -------------------- Euler update + per-WG ||dv||^2 partial (deterministic)
__global__ __launch_bounds__(256) void k_update(const float* __restrict__ parts,
                                                float* __restrict__ v,
                                                float* __restrict__ updp,
                                                const int* __restrict__ conv) {
  if (*conv) return;
  __shared__ float red[256];
  int i = blockIdx.x * 256 + threadIdx.x;
  float inst = 0.0f;
#pragma unroll
  for (int c = 0; c < NCHUNK; ++c) inst += parts[(size_t)c * (Bv * Mv) + i];
  float vo = v[i];
  float vn = vo + DT_TAU * (inst - vo);
  v[i] = vn;
  float d = vn - vo;
  red[threadIdx.x] = d * d;
  __syncthreads();
  for (int off = 128; off > 0; off >>= 1) {
    if (threadIdx.x < off) red[threadIdx.x] += red[threadIdx.x + off];
    __syncthreads();
  }
  if (threadIdx.x == 0) updp[blockIdx.x] = red[0];
}

__global__ __launch_bounds__(256) void k_check(const float* __restrict__ updp,
                                               int* __restrict__ conv) {
  if (*conv) return;
  __shared__ float red[256];
  red[threadIdx.x] = updp[threadIdx.x] + updp[threadIdx.x + 256];
  __syncthreads();
  for (int off = 128; off > 0; off >>= 1) {
    if (threadIdx.x < off) red[threadIdx.x] += red[threadIdx.x + off];
    __syncthreads();
  }
  if (threadIdx.x == 0 && sqrtf(red[0]) <= FP_THRES) *conv = 1;
}

__global__ __launch_bounds__(256) void k_copy(const float* __restrict__ v,
                                              float* __restrict__ out) {
  int i = blockIdx.x * 256 + threadIdx.x;
  out[i] = v[i];
}

// ---------------------------------------------------------------- launch
extern "C" void kernel_launch(void* const* d_in, const int* in_sizes, int n_in,
                              void* d_out, int out_size, void* d_ws, size_t ws_size,
                              hipStream_t stream) {
  const float* inp = (const float*)d_in[0];  // [64, 2048, 1]
  const float* W   = (const float*)d_in[1];  // [16384, 2048]
  float* out = (float*)d_out;

  float* ws    = (float*)d_ws;
  float* vbuf  = ws;                           // 131072
  float* Hbuf  = vbuf + (size_t)Bv * Mv;       // 1048576
  float* stats = Hbuf + (size_t)Bv * Nv;       // 128
  float* updp  = stats + 128;                  // 512
  int*   conv  = (int*)(updp + 512);           // 4 ints (1 used)
  float* parts = (float*)(conv + 4);           // NCHUNK * 131072

  k_init<<<512, 256, 0, stream>>>(inp, vbuf, conv);
  for (int s = 0; s < 50; ++s) {
    k_gemm1<<<dim3(Nv / 128), 256, 0, stream>>>(vbuf, W, Hbuf, conv);
    k_stats<<<dim3(Bv), 256, 0, stream>>>(Hbuf, stats, conv);
    k_gemm2<<<dim3(Mv / 128, NCHUNK), 256, 0, stream>>>(Hbuf, W, stats, parts, conv);
    k_update<<<512, 256, 0, stream>>>(parts, vbuf, updp, conv);
    k_check<<<1, 256, 0, stream>>>(updp, conv);
  }
  k_copy<<<512, 256, 0, stream>>>(vbuf, out);
}